// GatedGCNEdgesLayer_69269232550021
// MI455X (gfx1250) — compile-verified
//
#include <hip/hip_runtime.h>
#include <hip/hip_bf16.h>

#define DIM 128
#define BN_EPS 1e-5f
#define GATE_EPS 1e-6f

typedef float v2f __attribute__((ext_vector_type(2)));
typedef float v8f __attribute__((ext_vector_type(8)));

// ---------------------------------------------------------------------------
// Zero-fill (graph-capture-safe alternative to hipMemsetAsync)
// ---------------------------------------------------------------------------
__global__ void zero_kernel(float* __restrict__ p, long n) {
    long i = (long)blockIdx.x * blockDim.x + threadIdx.x;
    long stride = (long)gridDim.x * blockDim.x;
    for (; i < n; i += stride) p[i] = 0.0f;
}

// ---------------------------------------------------------------------------
// Fused 4-projection GEMM: out = h @ W + b for W in {Wa,Wb,Wd,We}.
// One block = one 16-row tile of h, staged ONCE in LDS and reused for all
// four weight matrices (4x less h traffic than one-matrix-per-block).
// 8 waves per block; wave w computes output columns [16w, 16w+16) via
// 32x V_WMMA_F32_16X16X4_F32 per matrix (K = 128), fp32 end to end.
// ---------------------------------------------------------------------------
__global__ __launch_bounds__(256) void gemm4_wmma_kernel(
    const float* __restrict__ h,
    const float* __restrict__ Wa, const float* __restrict__ ba,
    const float* __restrict__ Wb, const float* __restrict__ bb,
    const float* __restrict__ Wd, const float* __restrict__ bd,
    const float* __restrict__ We, const float* __restrict__ be,
    float* __restrict__ outA, float* __restrict__ outB,
    float* __restrict__ outD, float* __restrict__ outE,
    int nrows)
{
    __shared__ float lh[16][DIM];

    const int t    = threadIdx.x;
    const int row0 = blockIdx.x * 16;

    // Warm the weight matrices into the cache hierarchy (global_prefetch_b8)
    __builtin_prefetch(Wa, 0, 1);
    __builtin_prefetch(Wb, 0, 1);
    __builtin_prefetch(Wd, 0, 1);
    __builtin_prefetch(We, 0, 1);

    // Cooperative load of the 16x128 h tile (clamped, branch-free: EXEC full)
    #pragma unroll
    for (int i = 0; i < 8; ++i) {
        int idx = t + 256 * i;
        int r = idx >> 7;
        int c = idx & (DIM - 1);
        int gr = row0 + r;
        if (gr >= nrows) gr = nrows - 1;
        lh[r][c] = h[(long)gr * DIM + c];
    }
    __syncthreads();

    const int wave = t >> 5;            // 0..7 -> N tile
    const int lane = t & 31;
    const int l16  = lane & 15;
    const int koff = (lane >> 4) * 2;   // lanes 0-15: K0/K1, lanes 16-31: K2/K3
    const int ncol = wave * 16 + l16;
    const int mbase = row0 + ((lane >> 4) ? 8 : 0);

    for (int mat = 0; mat < 4; ++mat) {
        // Uniform scalar selection (no EXEC divergence)
        const float* W; const float* bias; float* out;
        switch (mat) {
            case 0:  W = Wa; bias = ba; out = outA; break;
            case 1:  W = Wb; bias = bb; out = outB; break;
            case 2:  W = Wd; bias = bd; out = outD; break;
            default: W = We; bias = be; out = outE; break;
        }

        v8f acc = {0.f, 0.f, 0.f, 0.f, 0.f, 0.f, 0.f, 0.f};
        #pragma unroll
        for (int kk = 0; kk < DIM / 4; ++kk) {
            const int k0 = kk * 4 + koff;
            v2f a, b;
            // A fragment (16x4 f32): row = l16, this lane's two K slices (LDS)
            a.x = lh[l16][k0];
            a.y = lh[l16][k0 + 1];
            // B fragment (4x16 f32): col = ncol, same K slices (weights, L2-hot)
            b.x = W[(long)k0 * DIM + ncol];
            b.y = W[(long)(k0 + 1) * DIM + ncol];
            acc = __builtin_amdgcn_wmma_f32_16x16x4_f32(
                /*neg_a=*/false, a, /*neg_b=*/false, b,
                /*c_mod=*/(short)0, acc, /*reuse_a=*/false, /*reuse_b=*/false);
        }

        // C/D layout: VGPR j -> M=j (lanes 0-15) / M=j+8 (lanes 16-31), N=lane%16
        const float bv = bias[ncol];
        #pragma unroll
        for (int j = 0; j < 8; ++j) {
            int gr = mbase + j;
            if (gr < nrows)
                out[(long)gr * DIM + ncol] = acc[j] + bv;
        }
    }
}

// ---------------------------------------------------------------------------
// Edge phase: sigma = sigmoid(Dh[src] + Eh[dst]);
// num[dst] += sigma * Bh[src]; den[dst] += sigma   (per-dim f32 atomics)
// 32 lanes per edge, float4 per lane; gathers are 512B rows that live in L2.
// ---------------------------------------------------------------------------
__global__ __launch_bounds__(256) void edge_kernel(
    const float* __restrict__ Bh, const float* __restrict__ Dh,
    const float* __restrict__ Eh,
    const int* __restrict__ src, const int* __restrict__ dst,
    float* __restrict__ num, float* __restrict__ den, int nedges)
{
    const int t = threadIdx.x;
    const int e = blockIdx.x * 8 + (t >> 5);
    if (e >= nedges) return;
    const int lane = t & 31;

    const int s = src[e];
    const int d = dst[e];
    const long so  = (long)s * DIM + lane * 4;
    const long dof = (long)d * DIM + lane * 4;

    const float4 dh = *(const float4*)(Dh + so);
    const float4 eh = *(const float4*)(Eh + dof);
    const float4 bh = *(const float4*)(Bh + so);

    const float x[4]  = {dh.x + eh.x, dh.y + eh.y, dh.z + eh.z, dh.w + eh.w};
    const float bv[4] = {bh.x, bh.y, bh.z, bh.w};
    #pragma unroll
    for (int i = 0; i < 4; ++i) {
        float sg = 1.0f / (1.0f + __expf(-x[i]));
        atomicAdd(num + dof + i, sg * bv[i]);
        atomicAdd(den + dof + i, sg);
    }
}

// ---------------------------------------------------------------------------
// h_new = Ah + num/(den+eps) (in place over Ah in d_out) and per-column
// sum / sum-of-squares partials for BatchNorm (register partials -> LDS ->
// one atomic per column per block).
// ---------------------------------------------------------------------------
__global__ __launch_bounds__(256) void hnew_stats_kernel(
    float* __restrict__ outh, const float* __restrict__ num,
    const float* __restrict__ den, float* __restrict__ gsum,
    float* __restrict__ gsq, int nrows)
{
    __shared__ float red[256];
    const int t    = threadIdx.x;
    const int col  = t & (DIM - 1);
    const int rp   = t >> 7;              // 0 or 1
    const int base = blockIdx.x * 32;

    float s = 0.0f, sq = 0.0f;
    #pragma unroll
    for (int i = 0; i < 16; ++i) {
        int r = base + rp + 2 * i;
        if (r < nrows) {
            long idx = (long)r * DIM + col;
            float v = outh[idx] + num[idx] / (den[idx] + GATE_EPS);
            outh[idx] = v;
            s  += v;
            sq += v * v;
        }
    }
    red[t] = s;
    __syncthreads();
    if (t < DIM) atomicAdd(gsum + t, red[t] + red[t + 128]);
    __syncthreads();
    red[t] = sq;
    __syncthreads();
    if (t < DIM) atomicAdd(gsq + t, red[t] + red[t + 128]);
}

// ---------------------------------------------------------------------------
// BatchNorm (biased batch stats) + ReLU, in place, float4 per thread.
// ---------------------------------------------------------------------------
__global__ __launch_bounds__(256) void bn_relu_kernel(
    float* __restrict__ outh, const float* __restrict__ gsum,
    const float* __restrict__ gsq, const float* __restrict__ gamma,
    const float* __restrict__ beta, int nrows)
{
    const long i4     = (long)blockIdx.x * 256 + threadIdx.x;
    const long total4 = (long)nrows * DIM / 4;
    if (i4 >= total4) return;

    const long base = i4 * 4;
    const int  col0 = (int)(base & (DIM - 1));   // aligned, no wrap within 4
    const float invN = 1.0f / (float)nrows;

    float4 v = *((const float4*)outh + i4);
    float vv[4] = {v.x, v.y, v.z, v.w};
    #pragma unroll
    for (int i = 0; i < 4; ++i) {
        int col = col0 + i;
        float mean = gsum[col] * invN;
        float var  = gsq[col] * invN - mean * mean;
        float y = (vv[i] - mean) * rsqrtf(var + BN_EPS) * gamma[col] + beta[col];
        vv[i] = y > 0.0f ? y : 0.0f;
    }
    float4 o = {vv[0], vv[1], vv[2], vv[3]};
    *((float4*)outh + i4) = o;
}

// ---------------------------------------------------------------------------
// Launch
// ---------------------------------------------------------------------------
extern "C" void kernel_launch(void* const* d_in, const int* in_sizes, int n_in,
                              void* d_out, int out_size, void* d_ws, size_t ws_size,
                              hipStream_t stream) {
    const float* h     = (const float*)d_in[0];
    const float* e     = (const float*)d_in[1];
    const int*   src   = (const int*)d_in[2];
    const int*   dst   = (const int*)d_in[3];
    const float* Wa    = (const float*)d_in[4];
    const float* ba    = (const float*)d_in[5];
    const float* Wb    = (const float*)d_in[6];
    const float* bb    = (const float*)d_in[7];
    const float* Wd    = (const float*)d_in[8];
    const float* bd    = (const float*)d_in[9];
    const float* We    = (const float*)d_in[10];
    const float* be    = (const float*)d_in[11];
    const float* gamma = (const float*)d_in[12];
    const float* beta  = (const float*)d_in[13];

    const int  N  = in_sizes[0] / DIM;
    const int  E  = in_sizes[1] / DIM;
    const long ND = (long)N * DIM;

    // Workspace layout (floats): Bh | Dh | Eh | num | den | gsum | gsq
    float* ws   = (float*)d_ws;
    float* Bh   = ws;
    float* Dh   = Bh + ND;
    float* Eh   = Dh + ND;
    float* num  = Eh + ND;
    float* den  = num + ND;
    float* gsum = den + ND;
    float* gsq  = gsum + DIM;

    float* outH = (float*)d_out;   // [N, DIM]; holds Ah first, then h_new
    float* outE = outH + ND;       // [E, DIM]; passthrough copy of e

    // e passthrough (D2D copy, graph-capture safe)
    hipMemcpyAsync(outE, e, (size_t)E * DIM * sizeof(float),
                   hipMemcpyDeviceToDevice, stream);

    // Zero accumulators: num, den, gsum, gsq are contiguous
    zero_kernel<<<1024, 256, 0, stream>>>(num, 2 * ND + 2 * DIM);

    // Fused 4x projection GEMM (WMMA f32 16x16x4); Ah -> d_out directly
    gemm4_wmma_kernel<<<(N + 15) / 16, 256, 0, stream>>>(
        h, Wa, ba, Wb, bb, Wd, bd, We, be, outH, Bh, Dh, Eh, N);

    // Edge gather / gate / scatter
    edge_kernel<<<(E + 7) / 8, 256, 0, stream>>>(Bh, Dh, Eh, src, dst, num, den, E);

    // h_new + BN statistics
    hnew_stats_kernel<<<(N + 31) / 32, 256, 0, stream>>>(outH, num, den, gsum, gsq, N);

    // BN normalize + ReLU
    bn_relu_kernel<<<(int)((ND / 4 + 255) / 256), 256, 0, stream>>>(
        outH, gsum, gsq, gamma, beta, N);
}